// Lya_Spec_76811195121797
// MI455X (gfx1250) — compile-verified
//
#include <hip/hip_runtime.h>
#include <stdint.h>

// Problem constants (from the reference setup).
#define T_STEPS 4096
#define BATCH   2048
#define CHUNK   8                      // matrices folded per chunk (accuracy bound: e^{(l1-l3)*C} << 1/eps_f32)
#define NCHUNK  (T_STEPS / CHUNK)      // 512
#define ROWS    (9 * CHUNK)            // 72 strided rows per tile
#define BW      128                    // batch columns per workgroup (tile width, contiguous)
#define TIME_DELTA 0.01f

#if defined(__HIP_DEVICE_COMPILE__) && __has_builtin(__builtin_amdgcn_tensor_load_to_lds)
#define HAVE_TDM 1
#else
#define HAVE_TDM 0
#endif

typedef unsigned int u32;
typedef __attribute__((ext_vector_type(4))) unsigned int u32x4;
typedef __attribute__((ext_vector_type(8))) int          i32x8;
typedef __attribute__((ext_vector_type(4))) int          i32x4;

// ---------------------------------------------------------------------------
// Phase 1: per-(chunk, b) 3x3 products P_c = J_{c*C+C-1} ... J_{c*C}.
// One workgroup = one chunk x 128 batch columns. The whole tile
// (72 rows of 128 contiguous f32, row stride = BATCH) is DMA'd into LDS by
// the Tensor Data Mover, then each thread folds its 8 matrices.
// ---------------------------------------------------------------------------
__global__ __launch_bounds__(BW) void lya_chunk_products(
    const float* __restrict__ J, float* __restrict__ P)
{
    __shared__ float tile[ROWS * BW];          // 72*128*4 = 36,864 bytes
    const int tid = threadIdx.x;
    const int c   = blockIdx.y;                // chunk index
    const int b0  = blockIdx.x * BW;           // batch-column base
    const float* src = J + (size_t)c * CHUNK * 9 * BATCH + b0;

#if HAVE_TDM
    if (tid < 32) {                            // one wave issues the TDM op
        const uint64_t ga  = (uint64_t)(uintptr_t)src;
        const u32      lds = (u32)(uintptr_t)(&tile[0]);   // flat low 32 bits == LDS offset
        u32x4 g0;
        g0.x = 1u;                                           // count=1 (valid user descriptor)
        g0.y = lds;                                          // lds_addr (bytes)
        g0.z = (u32)ga;                                      // global_addr[31:0]
        g0.w = (u32)((ga >> 32) & 0x01FFFFFFu) | (2u << 30); // global_addr[56:32] | type=2
        i32x8 g1;
        g1[0] = (2 << 16);        // workgroup_mask=0, data_size=2 (4 bytes), no pad/iterate
        g1[1] = (BW  << 16);      // atomic_barrier_addr=0 | tensor_dim0[15:0] = 128
        g1[2] = (ROWS << 16);     // tensor_dim0[31:16]=0 | tensor_dim1[15:0] = 72
        g1[3] = (BW  << 16);      // tensor_dim1[31:16]=0 | tile_dim0 = 128
        g1[4] = ROWS;             // tile_dim1 = 72 | tile_dim2 = 0 (2D tile)
        g1[5] = BATCH;            // tensor_dim0_stride[31:0] = 2048 elements
        g1[6] = 0;                // tensor_dim0_stride[47:32]=0 | tensor_dim1_stride lo = 0
        g1[7] = 0;
        i32x4 gz4 = {0, 0, 0, 0};             // groups 2/3 unused (<=3D tile)
        i32x8 gz8 = {0, 0, 0, 0, 0, 0, 0, 0}; // trailing group unused
        __builtin_amdgcn_tensor_load_to_lds(g0, g1, gz4, gz4, gz8, 0);
        __builtin_amdgcn_s_wait_tensorcnt(0);
    }
#else
    #pragma unroll 8
    for (int r = 0; r < ROWS; ++r)
        tile[r * BW + tid] = src[(size_t)r * BATCH + tid];
#endif
    __syncthreads();

    // Fold 8 row-major 3x3 matrices: P <- J_t * P
    float p[9] = {1.f, 0.f, 0.f, 0.f, 1.f, 0.f, 0.f, 0.f, 1.f};
    #pragma unroll
    for (int t = 0; t < CHUNK; ++t) {
        float j[9];
        #pragma unroll
        for (int k = 0; k < 9; ++k) j[k] = tile[(t * 9 + k) * BW + tid];
        float np[9];
        #pragma unroll
        for (int i = 0; i < 3; ++i)
            #pragma unroll
            for (int col = 0; col < 3; ++col)
                np[i * 3 + col] = j[i * 3 + 0] * p[0 * 3 + col]
                                + j[i * 3 + 1] * p[1 * 3 + col]
                                + j[i * 3 + 2] * p[2 * 3 + col];
        #pragma unroll
        for (int k = 0; k < 9; ++k) p[k] = np[k];
    }

    float* dst = P + (size_t)c * 9 * BATCH + b0 + tid;
    #pragma unroll
    for (int k = 0; k < 9; ++k) dst[(size_t)k * BATCH] = p[k];
}

// ---------------------------------------------------------------------------
// Phase 2: sequential scan over `steps` chunk products per batch column.
// M = P*Q, modified Gram-Schmidt, accumulate 0.5*ln(|beta_i|^2).
// Software-pipelined loads + global_prefetch_b8 lookahead (data is L2-hot).
// Also used as the exact C=1 fallback directly on the input.
// ---------------------------------------------------------------------------
__global__ __launch_bounds__(32) void lya_scan(
    const float* __restrict__ P, float* __restrict__ out, int steps)
{
    const int b = blockIdx.x * blockDim.x + threadIdx.x;
    if (b >= BATCH) return;

    float q[9] = {1.f, 0.f, 0.f, 0.f, 1.f, 0.f, 0.f, 0.f, 1.f};  // q[row*3+col]
    float ly0 = 0.f, ly1 = 0.f, ly2 = 0.f;

    float pn[9];
    #pragma unroll
    for (int k = 0; k < 9; ++k) pn[k] = P[(size_t)k * BATCH + b];

    for (int c = 0; c < steps; ++c) {
        float pc[9];
        #pragma unroll
        for (int k = 0; k < 9; ++k) pc[k] = pn[k];
        if (c + 1 < steps) {
            const float* nx = P + (size_t)(c + 1) * 9 * BATCH + b;
            #pragma unroll
            for (int k = 0; k < 9; ++k) pn[k] = nx[(size_t)k * BATCH];
        }
        if (c + 8 < steps)
            __builtin_prefetch(P + (size_t)(c + 8) * 9 * BATCH + b, 0, 0);

        // M = pc * q
        float m[9];
        #pragma unroll
        for (int i = 0; i < 3; ++i)
            #pragma unroll
            for (int jj = 0; jj < 3; ++jj)
                m[i * 3 + jj] = pc[i * 3 + 0] * q[0 * 3 + jj]
                              + pc[i * 3 + 1] * q[1 * 3 + jj]
                              + pc[i * 3 + 2] * q[2 * 3 + jj];

        // Modified Gram-Schmidt on columns of M
        float b00 = m[0], b01 = m[3], b02 = m[6];              // column 0
        float s0  = b00 * b00 + b01 * b01 + b02 * b02;
        float r0  = __builtin_amdgcn_rcpf(s0);

        float x0 = m[1], x1 = m[4], x2 = m[7];                 // column 1
        float a01 = b00 * x0 + b01 * x1 + b02 * x2;
        float t01 = a01 * r0;
        float b10 = x0 - t01 * b00, b11 = x1 - t01 * b01, b12 = x2 - t01 * b02;
        float s1  = b10 * b10 + b11 * b11 + b12 * b12;
        float r1  = __builtin_amdgcn_rcpf(s1);

        float y0 = m[2], y1 = m[5], y2 = m[8];                 // column 2
        float a02 = b00 * y0 + b01 * y1 + b02 * y2;
        float a12 = b10 * y0 + b11 * y1 + b12 * y2;
        float t02 = a02 * r0, t12 = a12 * r1;
        float b20 = y0 - t02 * b00 - t12 * b10;
        float b21 = y1 - t02 * b01 - t12 * b11;
        float b22 = y2 - t02 * b02 - t12 * b12;
        float s2  = b20 * b20 + b21 * b21 + b22 * b22;

        const float HALF_LN2 = 0.34657359027997264f;           // 0.5*ln(2)
        ly0 += HALF_LN2 * __builtin_amdgcn_logf(s0);           // 0.5*ln(s) via log2
        ly1 += HALF_LN2 * __builtin_amdgcn_logf(s1);
        ly2 += HALF_LN2 * __builtin_amdgcn_logf(s2);

        float in0 = __builtin_amdgcn_rsqf(s0);
        float in1 = __builtin_amdgcn_rsqf(s1);
        float in2 = __builtin_amdgcn_rsqf(s2);
        q[0] = b00 * in0; q[3] = b01 * in0; q[6] = b02 * in0;
        q[1] = b10 * in1; q[4] = b11 * in1; q[7] = b12 * in1;
        q[2] = b20 * in2; q[5] = b21 * in2; q[8] = b22 * in2;
    }

    const float inv = 1.0f / ((float)T_STEPS * TIME_DELTA);    // Lya = sum(log n)/ (T*dt)
    out[0 * BATCH + b] = ly0 * inv;
    out[1 * BATCH + b] = ly1 * inv;
    out[2 * BATCH + b] = ly2 * inv;
}

extern "C" void kernel_launch(void* const* d_in, const int* in_sizes, int n_in,
                              void* d_out, int out_size, void* d_ws, size_t ws_size,
                              hipStream_t stream) {
    (void)in_sizes; (void)n_in; (void)out_size;
    const float* J   = (const float*)d_in[0];
    float*       out = (float*)d_out;
    float*       ws  = (float*)d_ws;

    const size_t need = (size_t)NCHUNK * 9 * BATCH * sizeof(float);  // 37.7 MB
    if (ws_size >= need) {
        dim3 grid(BATCH / BW, NCHUNK);                // 16 x 512 workgroups
        lya_chunk_products<<<grid, BW, 0, stream>>>(J, ws);
        lya_scan<<<BATCH / 32, 32, 0, stream>>>(ws, out, NCHUNK);
    } else {
        // Exact (C=1) fallback: scan the raw Jacobians sequentially.
        lya_scan<<<BATCH / 32, 32, 0, stream>>>(J, out, T_STEPS);
    }
}